// SGCModule_83459804496279
// MI455X (gfx1250) — compile-verified
//
#include <hip/hip_runtime.h>
#include <stdint.h>

#define AS1 __attribute__((address_space(1)))
#define AS3 __attribute__((address_space(3)))

typedef int v4i __attribute__((ext_vector_type(4)));

#if __has_builtin(__builtin_amdgcn_global_load_async_to_lds_b128)
#define HAVE_ASYNC_BUILTIN 1
#else
#define HAVE_ASYNC_BUILTIN 0
#endif

#if __has_builtin(__builtin_amdgcn_s_wait_asynccnt)
#define WAIT_ASYNC(n) __builtin_amdgcn_s_wait_asynccnt(n)
#else
#define WAIT_ASYNC(n) asm volatile("s_wait_asynccnt %0" ::"i"(n) : "memory")
#endif

#define NBUF 4  // async pipeline depth: 3 tiles of gathers in flight + 1 consumed

// ---------------------------------------------------------------------------
// Codegen probe (never launched): shows at the top of the disasm snippet
// whether __hip_atomic_fetch_add(float, relaxed, agent) lowers to a native
// global_atomic_add_f32 or to a CAS loop on gfx1250.
// ---------------------------------------------------------------------------
extern "C" __global__ void atomic_probe(float* p, float v) {
  __hip_atomic_fetch_add(p, v, __ATOMIC_RELAXED, __HIP_MEMORY_SCOPE_AGENT);
}

// ---------------------------------------------------------------------------
// Gather-scale-scatter over edges. Tile = 16 edges (16 rows x 256B = 4KB)
// staged into LDS with GLOBAL_LOAD_ASYNC_TO_LDS_B128 (per-lane 16B random
// gather), 4-deep buffered on ASYNCcnt. Lane mapping: thread t -> edge (t>>4)
// of the tile, 16B chunk (t&15) of the 64-float row; producer lane ==
// consumer lane, so no barrier is needed. Scatter via fp32 hardware atomics
// (agent scope, relaxed).
// ---------------------------------------------------------------------------
__launch_bounds__(256, 4)
__global__ void sgc_scatter(const float4* __restrict__ xv,
                            const int* __restrict__ src,
                            const int* __restrict__ dst,
                            const float* __restrict__ rdeg,
                            float* __restrict__ out,
                            int n_edges, int tiles_total, int tiles_per_blk) {
  __shared__ float4 buf[NBUF][256];  // 4 x (16 edges x 16 chunks) = 16KB

  const int t = threadIdx.x;
  const int eL = t >> 4;   // edge within tile: 0..15
  const int ch = t & 15;   // float4 chunk within row: 0..15

  const int t0 = blockIdx.x * tiles_per_blk;
  int t1 = t0 + tiles_per_blk;
  if (t1 > tiles_total) t1 = tiles_total;
  if (t0 >= t1) return;

  auto issue = [&](int tile, int b) {
    int e = (tile << 4) + eL;
    if (e < n_edges) {
      int s = src[e];
      const float4* g = xv + (size_t)s * 16 + ch;
      v4i* l = (v4i*)&buf[b][(eL << 4) + ch];
#if HAVE_ASYNC_BUILTIN
      __builtin_amdgcn_global_load_async_to_lds_b128(
          (AS1 v4i*)(uintptr_t)g, (AS3 v4i*)l, 0, 0);
#else
      uint32_t lds_off = (uint32_t)(uintptr_t)(AS3 v4i*)l;
      asm volatile("global_load_async_to_lds_b128 %0, %1, off" ::"v"(lds_off),
                   "v"((uint64_t)(uintptr_t)g)
                   : "memory");
#endif
    }
  };

  // Prologue: fill the pipeline with NBUF-1 tiles of gathers.
#pragma unroll
  for (int k = 0; k < NBUF - 1; ++k) {
    if (t0 + k < t1) issue(t0 + k, k);
  }

  for (int tile = t0; tile < t1; ++tile) {
    const int b = (tile - t0) & (NBUF - 1);
    if (tile + NBUF - 1 < t1) {
      issue(tile + NBUF - 1, (tile + NBUF - 1 - t0) & (NBUF - 1));
      if (tile + 2 * NBUF < t1) {  // stream the edge-id arrays ahead
        __builtin_prefetch(&src[(tile + 2 * NBUF) << 4], 0, 0);
        __builtin_prefetch(&dst[(tile + 2 * NBUF) << 4], 0, 0);
      }
      WAIT_ASYNC(NBUF - 1);  // oldest in-flight tile (this one) has landed
    } else {
      WAIT_ASYNC(0);         // pipeline drain (last few tiles)
    }

    int e = (tile << 4) + eL;
    if (e < n_edges) {
      int s = src[e];
      int d = dst[e];
      float w = rdeg[s] * rdeg[d];  // 1/sqrt(deg_u * deg_v)
      float4 v = buf[b][(eL << 4) + ch];
      float* o = out + (size_t)d * 64 + (ch << 2);
      __hip_atomic_fetch_add(o + 0, v.x * w, __ATOMIC_RELAXED,
                             __HIP_MEMORY_SCOPE_AGENT);
      __hip_atomic_fetch_add(o + 1, v.y * w, __ATOMIC_RELAXED,
                             __HIP_MEMORY_SCOPE_AGENT);
      __hip_atomic_fetch_add(o + 2, v.z * w, __ATOMIC_RELAXED,
                             __HIP_MEMORY_SCOPE_AGENT);
      __hip_atomic_fetch_add(o + 3, v.w * w, __ATOMIC_RELAXED,
                             __HIP_MEMORY_SCOPE_AGENT);
    }
  }
}

// ---------------------------------------------------------------------------
// Zero the output accumulator and the degree scratch.
// ---------------------------------------------------------------------------
__global__ void sgc_zero(float* __restrict__ out, int n_out,
                         float* __restrict__ deg, int n_deg) {
  int i = blockIdx.x * blockDim.x + threadIdx.x;
  int stride = gridDim.x * blockDim.x;
  for (int k = i; k < n_out; k += stride) out[k] = 0.0f;
  for (int k = i; k < n_deg; k += stride) deg[k] = 0.0f;
}

// ---------------------------------------------------------------------------
// Out-degree histogram over src (fp32 counts, like the reference).
// ---------------------------------------------------------------------------
__global__ void sgc_degree(const int* __restrict__ src, int n_edges,
                           float* __restrict__ deg) {
  int e = blockIdx.x * blockDim.x + threadIdx.x;
  if (e < n_edges) {
    __hip_atomic_fetch_add(&deg[src[e]], 1.0f, __ATOMIC_RELAXED,
                           __HIP_MEMORY_SCOPE_AGENT);
  }
}

// ---------------------------------------------------------------------------
// deg[i] <- rsqrt(deg[i])  (zero degree -> +inf, matching 1/sqrt(0)).
// ---------------------------------------------------------------------------
__global__ void sgc_rsqrt(float* __restrict__ deg, int n_nodes) {
  int i = blockIdx.x * blockDim.x + threadIdx.x;
  if (i < n_nodes) deg[i] = rsqrtf(deg[i]);
}

// ---------------------------------------------------------------------------
// Launcher
// ---------------------------------------------------------------------------
extern "C" void kernel_launch(void* const* d_in, const int* in_sizes, int n_in,
                              void* d_out, int out_size, void* d_ws,
                              size_t ws_size, hipStream_t stream) {
  const float* x = (const float*)d_in[0];
  const int* src = (const int*)d_in[1];
  const int* dst = (const int*)d_in[2];
  float* out = (float*)d_out;

  const int n_edges = in_sizes[1];
  const int n_nodes = in_sizes[0] / 64;

  float* deg = (float*)d_ws;  // n_nodes floats of scratch

  sgc_zero<<<512, 256, 0, stream>>>(out, n_nodes * 64, deg, n_nodes);
  sgc_degree<<<(n_edges + 255) / 256, 256, 0, stream>>>(src, n_edges, deg);
  sgc_rsqrt<<<(n_nodes + 255) / 256, 256, 0, stream>>>(deg, n_nodes);

  const int tiles = (n_edges + 15) / 16;
  int blocks = tiles < 1024 ? tiles : 1024;
  int tiles_per_blk = (tiles + blocks - 1) / blocks;
  sgc_scatter<<<blocks, 256, 0, stream>>>((const float4*)x, src, dst, deg, out,
                                          n_edges, tiles, tiles_per_blk);
}